// SelfHistorySelector_57896159150302
// MI455X (gfx1250) — compile-verified
//
#include <hip/hip_runtime.h>
#include <math.h>

#define B_ 2048
#define T_ 200
#define H_ 256
#define K_ 8

typedef __attribute__((ext_vector_type(2))) float v2f;
typedef __attribute__((ext_vector_type(4))) float v4f;
typedef __attribute__((ext_vector_type(8))) float v8f;

// ---------------------------------------------------------------------------
// Kernel 1: per 16-row stripe of batch:
//   Q  = current_state @ Wq + bq            (16 x 256, kept in LDS)
//   cb = bk . q                              (16)
//   R  = Q @ Wk^T                            (16 x 256, to workspace)
// Uses V_WMMA_F32_16X16X4_F32 (fp32 in, fp32 accumulate -> full precision).
// ---------------------------------------------------------------------------
__global__ __launch_bounds__(256)
void shs_qr_kernel(const float* __restrict__ cs, const float* __restrict__ Wq,
                   const float* __restrict__ bq, const float* __restrict__ Wk,
                   const float* __restrict__ bk, float* __restrict__ R,
                   float* __restrict__ cb) {
  __shared__ float qld[16 * H_];
  const int m0   = blockIdx.x * 16;
  const int lane = threadIdx.x & 31;
  const int wave = threadIdx.x >> 5;
  const int lh   = lane & 15;          // M (for A) / N (for B) within tile
  const bool hi  = lane >= 16;         // lanes 16-31 carry K+2,K+3

#if __has_builtin(__builtin_amdgcn_wmma_f32_16x16x4_f32)
  // ---- Pass 1: Q stripe ----
  for (int tile = 0; tile < 2; ++tile) {
    const int n0 = wave * 32 + tile * 16;
    v8f acc = {};
    for (int k0 = 0; k0 < H_; k0 += 4) {
      const int ka = k0 + (hi ? 2 : 0);
      v2f a, b;
      a.x = cs[(m0 + lh) * H_ + ka];        // A[M=lh][K=ka]
      a.y = cs[(m0 + lh) * H_ + ka + 1];
      b.x = Wq[ka * H_ + n0 + lh];          // B[K=ka][N=n0+lh]
      b.y = Wq[(ka + 1) * H_ + n0 + lh];
      acc = __builtin_amdgcn_wmma_f32_16x16x4_f32(false, a, false, b,
                                                  (short)0, acc, false, false);
    }
    const float bias = bq[n0 + lh];
#pragma unroll
    for (int v = 0; v < 8; ++v) {
      const int m = v + (hi ? 8 : 0);       // C/D: vgpr v -> M=v (lanes 0-15), M=v+8 (16-31)
      qld[m * H_ + n0 + lh] = acc[v] + bias;
    }
  }
  __syncthreads();
  if (threadIdx.x < 16) {
    float s = 0.0f;
    for (int h = 0; h < H_; ++h) s += bk[h] * qld[threadIdx.x * H_ + h];
    cb[m0 + threadIdx.x] = s;
  }
  // ---- Pass 2: R = Q @ Wk^T  (B'[k=h][n=j] = Wk[j][h]) ----
  for (int tile = 0; tile < 2; ++tile) {
    const int n0 = wave * 32 + tile * 16;
    v8f acc = {};
    for (int h0 = 0; h0 < H_; h0 += 4) {
      const int ka = h0 + (hi ? 2 : 0);
      v2f a, b;
      a.x = qld[lh * H_ + ka];
      a.y = qld[lh * H_ + ka + 1];
      b.x = Wk[(n0 + lh) * H_ + ka];        // Wk[j][h] with j = n0+lh, h = ka
      b.y = Wk[(n0 + lh) * H_ + ka + 1];
      acc = __builtin_amdgcn_wmma_f32_16x16x4_f32(false, a, false, b,
                                                  (short)0, acc, false, false);
    }
#pragma unroll
    for (int v = 0; v < 8; ++v) {
      const int m = v + (hi ? 8 : 0);
      R[(m0 + m) * H_ + n0 + lh] = acc[v];
    }
  }
#else
  // Scalar fallback (keeps compile alive if the fp32 WMMA builtin differs)
  for (int m = 0; m < 16; ++m) {
    float s = 0.0f;
    for (int k = 0; k < H_; ++k) s += cs[(m0 + m) * H_ + k] * Wq[k * H_ + threadIdx.x];
    qld[m * H_ + threadIdx.x] = s + bq[threadIdx.x];
  }
  __syncthreads();
  if (threadIdx.x < 16) {
    float s = 0.0f;
    for (int h = 0; h < H_; ++h) s += bk[h] * qld[threadIdx.x * H_ + h];
    cb[m0 + threadIdx.x] = s;
  }
  for (int m = 0; m < 16; ++m) {
    float s = 0.0f;
    for (int h = 0; h < H_; ++h) s += qld[m * H_ + h] * Wk[threadIdx.x * H_ + h];
    R[(m0 + m) * H_ + threadIdx.x] = s;
  }
#endif
}

// ---------------------------------------------------------------------------
// Kernel 2: one workgroup per batch row b.
//   scores[t] = (S[b,t,:].r_b + bk.q_b) / 16   (streaming, memory-bound, NT)
//   masked softmax -> top-8 sparsify -> renorm -> attn, selected
//   summary = sum_k w_k * relu(S[b,t_k,:] @ Wv + bv)   (only 8 rows!)
// ---------------------------------------------------------------------------
__device__ __forceinline__ v4f nt_load4(const float* p) {
  return __builtin_nontemporal_load((const v4f*)p);
}

__global__ __launch_bounds__(256)
void shs_attn_kernel(const float* __restrict__ S, const int* __restrict__ vm,
                     const float* __restrict__ Wv, const float* __restrict__ bvp,
                     const float* __restrict__ R, const float* __restrict__ cb,
                     float* __restrict__ out_summary, float* __restrict__ out_attn,
                     int* __restrict__ out_sel) {
  __shared__ float r_sh[H_];
  __shared__ float sc[T_];
  __shared__ int   hm[T_];
  __shared__ float redf[256];
  __shared__ int   redi[256];
  __shared__ float selw[K_];
  __shared__ int   seli[K_];
  __shared__ int   selo[K_];
  __shared__ float rows[K_ * H_];

  const int b    = blockIdx.x;
  const int tid  = threadIdx.x;
  const int lane = tid & 31;
  const int wave = tid >> 5;

  r_sh[tid] = R[(size_t)b * H_ + tid];

  // valid_counts = sum(mask) ; history_mask = mask & (t < valid-1)
  int mval = 0;
  if (tid < T_) mval = (vm[(size_t)b * T_ + tid] != 0) ? 1 : 0;
  redi[tid] = mval;
  __syncthreads();
  for (int s = 128; s > 0; s >>= 1) {
    if (tid < s) redi[tid] += redi[tid + s];
    __syncthreads();
  }
  const int valid = redi[0];
  __syncthreads();
  if (tid < T_) hm[tid] = (mval && (tid < valid - 1)) ? 1 : 0;
  __syncthreads();

  // ---- scores: each wave streams rows; lane-invariant r fragment in VGPRs ----
  const float cbv = cb[b];
  const float r0 = r_sh[lane * 4 + 0], r1 = r_sh[lane * 4 + 1];
  const float r2 = r_sh[lane * 4 + 2], r3 = r_sh[lane * 4 + 3];
  const float r4 = r_sh[128 + lane * 4 + 0], r5 = r_sh[128 + lane * 4 + 1];
  const float r6 = r_sh[128 + lane * 4 + 2], r7 = r_sh[128 + lane * 4 + 3];

  for (int t = wave; t < T_; t += 16) {
    const int t2 = t + 8;
    const bool have2 = t2 < T_;
    const float* rowA = S + ((size_t)b * T_ + t) * (size_t)H_;
    const float* rowB = S + ((size_t)b * T_ + (have2 ? t2 : t)) * (size_t)H_;
    // issue all 4 b128 NT loads up front for MLP
    v4f a0 = nt_load4(rowA + lane * 4);
    v4f a1 = nt_load4(rowA + 128 + lane * 4);
    v4f b0 = nt_load4(rowB + lane * 4);
    v4f b1 = nt_load4(rowB + 128 + lane * 4);

    float dA = a0.x * r0 + a0.y * r1 + a0.z * r2 + a0.w * r3 +
               a1.x * r4 + a1.y * r5 + a1.z * r6 + a1.w * r7;
    float dB = b0.x * r0 + b0.y * r1 + b0.z * r2 + b0.w * r3 +
               b1.x * r4 + b1.y * r5 + b1.z * r6 + b1.w * r7;
#pragma unroll
    for (int off = 16; off > 0; off >>= 1) {
      dA += __shfl_xor(dA, off, 32);
      dB += __shfl_xor(dB, off, 32);
    }
    if (lane == 0) {
      sc[t] = hm[t] ? (dA + cbv) * 0.0625f : -1e9f;
      if (have2) sc[t2] = hm[t2] ? (dB + cbv) * 0.0625f : -1e9f;
    }
  }
  __syncthreads();

  // ---- masked softmax + renorm (matches reference semantics) ----
  float myv = (tid < T_) ? sc[tid] : -3.0e38f;
  redf[tid] = myv;
  __syncthreads();
  for (int s = 128; s > 0; s >>= 1) {
    if (tid < s) redf[tid] = fmaxf(redf[tid], redf[tid + s]);
    __syncthreads();
  }
  const float mx = redf[0];
  __syncthreads();
  float e = (tid < T_ && hm[tid]) ? __expf(myv - mx) : 0.0f;
  redf[tid] = e;
  __syncthreads();
  for (int s = 128; s > 0; s >>= 1) {
    if (tid < s) redf[tid] += redf[tid + s];
    __syncthreads();
  }
  const float ssum = redf[0];
  __syncthreads();
  const float inv = (ssum > 0.0f) ? (1.0f / ssum) : 0.0f;
  if (tid < T_) sc[tid] = hm[tid] ? (e * inv) : -1.0f;  // masked attn for top-k
  __syncthreads();

  // ---- top-8 (value desc, index asc tie-break) — wave 0 only ----
  for (int k = 0; k < K_; ++k) {
    if (wave == 0) {
      float bv = -3.0f;
      int bi = T_;
      for (int t = lane; t < T_; t += 32) {
        const float c = sc[t];
        if (c > bv) { bv = c; bi = t; }   // ascending scan keeps lowest idx on tie
      }
#pragma unroll
      for (int off = 16; off > 0; off >>= 1) {
        const float ov = __shfl_xor(bv, off, 32);
        const int   oi = __shfl_xor(bi, off, 32);
        if (ov > bv || (ov == bv && oi < bi)) { bv = ov; bi = oi; }
      }
      if (lane == 0) { seli[k] = bi; selw[k] = bv; sc[bi] = -2.0f; }
    }
    __syncthreads();
  }

  if (tid == 0) {
    float tot = 0.0f;
    for (int k = 0; k < K_; ++k) tot += fmaxf(selw[k], 0.0f);
    const float den = fmaxf(tot, 1.1920929e-07f);  // float32 eps
    for (int k = 0; k < K_; ++k) {
      const bool ok = selw[k] >= 0.0f;             // -1 marks non-history slot
      selo[k] = ok ? seli[k] : -1;
      selw[k] = ok ? (fmaxf(selw[k], 0.0f) / den) : 0.0f;
    }
  }
  __syncthreads();

  if (tid < K_) out_sel[(size_t)b * K_ + tid] = selo[tid];

  // dense sparse-attn output
  for (int t = tid; t < T_; t += 256) {
    float w = 0.0f;
#pragma unroll
    for (int k = 0; k < K_; ++k)
      if (seli[k] == t && selo[k] >= 0) w = selw[k];
    out_attn[(size_t)b * T_ + t] = w;
  }

  // ---- gather the 8 selected rows, compute summary ----
#pragma unroll
  for (int k = 0; k < K_; ++k) {
    const int t = seli[k];  // always < T_; contributes 0 if selw[k]==0
    rows[k * H_ + tid] = S[((size_t)b * T_ + t) * (size_t)H_ + tid];
  }
  __syncthreads();

  float acc[K_];
#pragma unroll
  for (int k = 0; k < K_; ++k) acc[k] = 0.0f;
  for (int j = 0; j < H_; ++j) {
    const float wv = Wv[(size_t)j * H_ + tid];  // coalesced; Wv stays in L2
#pragma unroll
    for (int k = 0; k < K_; ++k) acc[k] = fmaf(rows[k * H_ + j], wv, acc[k]);
  }
  const float bb = bvp[tid];
  float sum = 0.0f;
#pragma unroll
  for (int k = 0; k < K_; ++k) sum += selw[k] * fmaxf(acc[k] + bb, 0.0f);
  out_summary[(size_t)b * H_ + tid] = sum;  // all-w==0 (no history) -> 0
}

extern "C" void kernel_launch(void* const* d_in, const int* in_sizes, int n_in,
                              void* d_out, int out_size, void* d_ws, size_t ws_size,
                              hipStream_t stream) {
  const float* cs = (const float*)d_in[0];   // current_state (B,H)
  const float* S  = (const float*)d_in[1];   // state_sequence (B,T,H)
  const int*   vm = (const int*)d_in[2];     // visit_mask (B,T)
  const float* Wq = (const float*)d_in[3];
  const float* bq = (const float*)d_in[4];
  const float* Wk = (const float*)d_in[5];
  const float* bk = (const float*)d_in[6];
  const float* Wv = (const float*)d_in[7];
  const float* bv = (const float*)d_in[8];

  float* ws = (float*)d_ws;
  float* R  = ws;                    // B*H floats
  float* cb = ws + (size_t)B_ * H_;  // B floats

  float* out         = (float*)d_out;
  float* out_summary = out;                                   // B*H
  float* out_attn    = out + (size_t)B_ * H_;                 // B*T
  int*   out_sel     = (int*)(out + (size_t)B_ * H_ + (size_t)B_ * T_);  // B*K

  shs_qr_kernel<<<B_ / 16, 256, 0, stream>>>(cs, Wq, bq, Wk, bk, R, cb);
  shs_attn_kernel<<<B_, 256, 0, stream>>>(S, vm, Wv, bv, R, cb,
                                          out_summary, out_attn, out_sel);
}